// SKANLinear_59313498358108
// MI455X (gfx1250) — compile-verified
//
#include <hip/hip_runtime.h>

// SKANLinear forward on MI455X (gfx1250, wave32, WMMA + async-LDS copies).
//
//   y[m,n] = 0.5 * ( sum_k x[m,k]*w[n,k] + sum_k |x[m,k]|*|w[n,k]| ) + relu(w[n,2048])
//
// Compute-bound (AI ~ 900 FLOP/B): exact-precision V_WMMA_F32_16X16X4_F32 core,
// double-buffered LDS tiles filled with GLOBAL_LOAD_ASYNC_TO_LDS (ASYNCcnt),
// so global traffic overlaps the matrix pipe instead of stalling at barriers.

typedef __attribute__((ext_vector_type(2))) float v2f;
typedef __attribute__((ext_vector_type(8))) float v8f;

#define M_DIM 8192   // 4*2048 rows after flatten
#define N_DIM 2048   // out_features
#define K_DIM 2048   // in_features (bias column handled in epilogue)
#define LDW   2049   // weight row stride (in_features + 1)

#define BM 128
#define BN 128
#define BK 16
#define TLD 20       // LDS tile row stride in floats (16B-aligned rows, conflict-free)

__device__ __forceinline__ void async_ld_b128(unsigned lds_byte, int gbyte_off,
                                              const float* base) {
    // GVS mode: mem = SGPR64 + VGPR_i32 (byte) ; VDST = LDS byte address
    asm volatile("global_load_async_to_lds_b128 %0, %1, %2"
                 :: "v"(lds_byte), "v"(gbyte_off), "s"(base) : "memory");
}
__device__ __forceinline__ void async_ld_b32(unsigned lds_byte, int gbyte_off,
                                             const float* base) {
    asm volatile("global_load_async_to_lds_b32 %0, %1, %2"
                 :: "v"(lds_byte), "v"(gbyte_off), "s"(base) : "memory");
}
__device__ __forceinline__ void wait_async0() {
    asm volatile("s_wait_asynccnt 0" ::: "memory");
}

__global__ __launch_bounds__(256)
void skan_wmma_f32(const float* __restrict__ x,
                   const float* __restrict__ w,
                   float* __restrict__ y)
{
    __shared__ float As[2][BM * TLD];   // A tiles: [m][k], 128x16 (+pad), ping-pong
    __shared__ float Bs[2][BN * TLD];   // B tiles: [n][k] (weight is [out][in])

    const int tid  = threadIdx.x;
    const int wave = tid >> 5;          // 8 waves of 32
    const int lane = tid & 31;
    const int hi   = lane >> 4;         // half-wave select (ISA fragment layout)
    const int l16  = lane & 15;

    const int wm = wave >> 1;           // 0..3 : 32-row wave tile in M
    const int wn = wave & 1;            // 0..1 : 64-col wave tile in N

    const int m0 = blockIdx.y * BM;
    const int n0 = blockIdx.x * BN;

    // LDS byte base addresses (generic-pointer low 32 bits == LDS byte offset)
    const unsigned as_base = (unsigned)(uintptr_t)(&As[0][0]);
    const unsigned bs_base = (unsigned)(uintptr_t)(&Bs[0][0]);
    const unsigned abuf_sz = (unsigned)(BM * TLD * 4);
    const unsigned bbuf_sz = (unsigned)(BN * TLD * 4);

    // Per-thread staging positions (constant across K-steps except +k0*4)
    const int arow0 = tid >> 2;                 // A: two b128 per thread
    const int ac4   = (tid & 3) << 2;
    const int arow1 = (tid + 256) >> 2;
    const int brow  = tid >> 1;                 // B: eight b32 per thread
    const int bcs   = (tid & 1) << 3;

    auto load_tile = [&](int k0, int buf) {
        const unsigned ab = as_base + (unsigned)buf * abuf_sz;
        const unsigned bb = bs_base + (unsigned)buf * bbuf_sz;
        // A tile: x rows are 16B aligned at k0 multiples of 16 -> b128 async
        async_ld_b128(ab + (unsigned)((arow0 * TLD + ac4) << 2),
                      ((m0 + arow0) * K_DIM + k0 + ac4) << 2, x);
        async_ld_b128(ab + (unsigned)((arow1 * TLD + ac4) << 2),
                      ((m0 + arow1) * K_DIM + k0 + ac4) << 2, x);
        // B tile: ldw = 2049 (odd) -> rows only 4B aligned -> b32 async
        #pragma unroll
        for (int j = 0; j < 8; ++j)
            async_ld_b32(bb + (unsigned)((brow * TLD + bcs + j) << 2),
                         ((n0 + brow) * LDW + k0 + bcs + j) << 2, w);
    };

    v8f acc_s[2][4];                    // signed term accumulators
    v8f acc_a[2][4];                    // abs term accumulators
    #pragma unroll
    for (int i = 0; i < 2; ++i)
        #pragma unroll
        for (int j = 0; j < 4; ++j) { acc_s[i][j] = (v8f){}; acc_a[i][j] = (v8f){}; }

    // Prologue: prefetch tile 0
    load_tile(0, 0);
    wait_async0();
    __syncthreads();

    for (int k0 = 0; k0 < K_DIM; k0 += BK) {
        const int buf = (k0 / BK) & 1;

        // Kick off next tile into the other buffer (overlaps with WMMA below)
        if (k0 + BK < K_DIM) load_tile(k0 + BK, buf ^ 1);

        const float* At = &As[buf][0];
        const float* Bt = &Bs[buf][0];

        // ---- compute: 4 K-slabs of 4, 2x4 tiles, signed + abs ----
        #pragma unroll
        for (int kk = 0; kk < BK; kk += 4) {
            v2f a[2], aa[2], b[4], ba[4];
            #pragma unroll
            for (int mt = 0; mt < 2; ++mt) {
                int m = wm * 32 + mt * 16 + l16;
                a[mt] = *(const v2f*)(At + m * TLD + kk + 2 * hi);
                aa[mt][0] = __builtin_fabsf(a[mt][0]);
                aa[mt][1] = __builtin_fabsf(a[mt][1]);
            }
            #pragma unroll
            for (int nt = 0; nt < 4; ++nt) {
                int n = wn * 64 + nt * 16 + l16;
                b[nt] = *(const v2f*)(Bt + n * TLD + kk + 2 * hi);
                ba[nt][0] = __builtin_fabsf(b[nt][0]);
                ba[nt][1] = __builtin_fabsf(b[nt][1]);
            }
            #pragma unroll
            for (int mt = 0; mt < 2; ++mt)
                #pragma unroll
                for (int nt = 0; nt < 4; ++nt) {
                    acc_s[mt][nt] = __builtin_amdgcn_wmma_f32_16x16x4_f32(
                        false, a[mt], false, b[nt], (short)0, acc_s[mt][nt], false, false);
                    acc_a[mt][nt] = __builtin_amdgcn_wmma_f32_16x16x4_f32(
                        false, aa[mt], false, ba[nt], (short)0, acc_a[mt][nt], false, false);
                }
        }

        // Retire our async writes for tile i+1, then barrier: after it, all
        // waves are done reading buf i and all writes to buf i^1 are visible.
        wait_async0();
        __syncthreads();
    }

    // ---- epilogue: combine terms, add relu(bias weight), store per C/D layout ----
    #pragma unroll
    for (int nt = 0; nt < 4; ++nt) {
        int n = n0 + wn * 64 + nt * 16 + l16;
        float wb   = w[(size_t)n * LDW + K_DIM];     // weight[:, in_features] (bias col)
        float bias = wb > 0.0f ? wb : 0.0f;          // 0.5*(wb + |wb|) == relu(wb)
        #pragma unroll
        for (int mt = 0; mt < 2; ++mt) {
            int mbase = m0 + wm * 32 + mt * 16 + hi * 8;
            #pragma unroll
            for (int r = 0; r < 8; ++r) {
                float v = 0.5f * (acc_s[mt][nt][r] + acc_a[mt][nt][r]) + bias;
                y[(size_t)(mbase + r) * N_DIM + n] = v;
            }
        }
    }
}

extern "C" void kernel_launch(void* const* d_in, const int* in_sizes, int n_in,
                              void* d_out, int out_size, void* d_ws, size_t ws_size,
                              hipStream_t stream) {
    (void)in_sizes; (void)n_in; (void)out_size; (void)d_ws; (void)ws_size;
    const float* x = (const float*)d_in[0];   // (4,2048,2048) f32
    const float* w = (const float*)d_in[1];   // (2048,2049)  f32
    float* y = (float*)d_out;                 // (8192,2048)  f32

    dim3 grid(N_DIM / BN, M_DIM / BM);        // (16, 64)
    skan_wmma_f32<<<grid, dim3(256), 0, stream>>>(x, w, y);
}